// TLAlw_40209483825862
// MI455X (gfx1250) — compile-verified
//
#include <hip/hip_runtime.h>
#include <stdint.h>

// Sliding-window (W=64) minimum along T per batch row.
//   y[b,t] = min(x[b, max(0,t-63) .. t])
// B=4096, T=8192, f32. Pure bandwidth problem: 256 MiB traffic -> ~11us @23.3TB/s.
// Strategy: one workgroup per row; async-DMA the row into LDS, block prefix/suffix
// min trick (3 v_min per element), async-DMA results back out of LDS.

#define T_LEN     8192
#define W         64
#define NBLK      (T_LEN / W)      // 128 blocks of 64 along T
#define NTHREADS  NBLK             // one thread per output block (4 wave32)
#define PAD_ROW   68               // 64 data dwords + 4 pad: 272B stride keeps 16B align
#define LDS_DW    (NBLK * PAD_ROW) // 8704 dwords = 34816 bytes
#define NCHUNK    (T_LEN / 4)      // 2048 x b128 chunks per row
#define CHUNKS_PER_THREAD (NCHUNK / NTHREADS) // 16
#define INIT_VALF 9999.0f

__device__ __forceinline__ uint32_t lds_off(const void* p) {
  // addrspacecast(AS3 -> generic) puts the allocation-relative LDS byte offset
  // in the low 32 bits; async-LDS instructions take exactly that in a VGPR.
  return (uint32_t)(uintptr_t)p;
}

__global__ __launch_bounds__(NTHREADS)
void TLAlw_window_min_kernel(const float* __restrict__ x,
                             float* __restrict__ y) {
  __shared__ float lds[LDS_DW];

  const int j   = threadIdx.x;              // block index within row, 0..127
  const size_t row = (size_t)blockIdx.x * T_LEN;

  // ---------------- Phase 1: async load row -> LDS (padded layout) ----------
  // chunk c covers t = 4c..4c+3 ; LDS dword = (c/16)*PAD_ROW + (c%16)*4
#pragma unroll
  for (int k = 0; k < CHUNKS_PER_THREAD; ++k) {
    const int c = j + k * NTHREADS;         // consecutive chunks across lanes
    const uint64_t ga = (uint64_t)(uintptr_t)(x + row + (size_t)c * 4);
    const uint32_t la = lds_off(&lds[(c >> 4) * PAD_ROW + (c & 15) * 4]);
    asm volatile("global_load_async_to_lds_b128 %0, %1, off"
                 :: "v"(la), "v"(ga) : "memory");
  }
  asm volatile("s_wait_asynccnt 0" ::: "memory");
  __syncthreads();

  // ---------------- Phase 2: per-block suffix/prefix min --------------------
  // s[i] = min(prevBlock[i..63]); for block 0 the "previous" data is INIT_VAL.
  float s[W];
  if (j > 0) {
    const int pb = (j - 1) * PAD_ROW;
    float run = INIT_VALF;
#pragma unroll
    for (int i = W - 1; i >= 1; --i) {
      run = fminf(run, lds[pb + i]);
      s[i] = run;
    }
  } else {
#pragma unroll
    for (int i = 1; i < W; ++i) s[i] = INIT_VALF;
  }

  float out[W];
  {
    const int cb = j * PAD_ROW;
    float p = INIT_VALF;
#pragma unroll
    for (int t = 0; t < W; ++t) {
      p = fminf(p, lds[cb + t]);
      out[t] = (t < W - 1) ? fminf(p, s[t + 1]) : p;
    }
  }

  // All cross-thread LDS reads are done; reuse the buffer for the output.
  __syncthreads();
  {
    const int cb = j * PAD_ROW;
#pragma unroll
    for (int t = 0; t < W; ++t) lds[cb + t] = out[t];
  }
  __syncthreads();

  // ---------------- Phase 3: async store LDS -> global ----------------------
#pragma unroll
  for (int k = 0; k < CHUNKS_PER_THREAD; ++k) {
    const int c = j + k * NTHREADS;
    const uint64_t ga = (uint64_t)(uintptr_t)(y + row + (size_t)c * 4);
    const uint32_t la = lds_off(&lds[(c >> 4) * PAD_ROW + (c & 15) * 4]);
    asm volatile("global_store_async_from_lds_b128 %0, %1, off"
                 :: "v"(ga), "v"(la) : "memory");
  }
  // S_ENDPGM performs an implicit wait-idle, but drain explicitly before the
  // workgroup (and its LDS allocation) retires.
  asm volatile("s_wait_asynccnt 0" ::: "memory");
}

extern "C" void kernel_launch(void* const* d_in, const int* in_sizes, int n_in,
                              void* d_out, int out_size, void* d_ws, size_t ws_size,
                              hipStream_t stream) {
  const float* x = (const float*)d_in[0];
  float* y = (float*)d_out;
  const int B = in_sizes[0] / T_LEN;   // 4096 for the reference shapes
  TLAlw_window_min_kernel<<<B, NTHREADS, 0, stream>>>(x, y);
  (void)n_in; (void)out_size; (void)d_ws; (void)ws_size;
}